// RelationalGraphConvolution_42391327212275
// MI455X (gfx1250) — compile-verified
//
#include <hip/hip_runtime.h>

#define D_DIM 128
#define O_DIM 128

typedef float v2f __attribute__((ext_vector_type(2)));
typedef float v8f __attribute__((ext_vector_type(8)));

// ---------------------------------------------------------------------------
// Dual dense projection: Y1 = X @ W1, Y2 = X @ W2   (fp32 WMMA 16x16x4)
// One 16-row tile per block (blockIdx.x), 8 waves per block, wave w owns
// output-column tile n0 = 16*w (O = 128 = 8 * 16).
// A-fragment (rows of X) is loaded once per k-step and fed to BOTH WMMAs.
// Epilogue guard is uniform per tile: full tiles take a branchless 16-store
// path; only the (rare) tail tile pays per-row EXEC masking.
// ---------------------------------------------------------------------------
__global__ void __launch_bounds__(256)
rgcn_gemm_dual(const float* __restrict__ X,
               const float* __restrict__ W1,
               const float* __restrict__ W2,
               float* __restrict__ Y1,
               float* __restrict__ Y2,
               int nRows)
{
    const int lane = threadIdx.x & 31;
    const int wave = threadIdx.x >> 5;
    const int half = lane >> 4;        // 0: lanes 0-15, 1: lanes 16-31
    const int l    = lane & 15;

    const int row0 = blockIdx.x * 16;
    const int n0   = wave * 16;

    // Clamp A row for loads (branchless, keeps EXEC uniform for WMMA).
    int arowIdx = row0 + l;
    if (arowIdx > nRows - 1) arowIdx = nRows - 1;
    const float* __restrict__ arow = X + (size_t)arowIdx * D_DIM;

    v8f c1 = {};
    v8f c2 = {};

#pragma unroll 4
    for (int k = 0; k < D_DIM; k += 4) {
        const int ka = k + 2 * half;           // K position for this half-wave
        // A fragment: A[m][ka], A[m][ka+1]  -> single b64 load
        v2f a = *(const v2f*)(arow + ka);
        // B fragments: B[ka][n], B[ka+1][n]
        const size_t bi0 = (size_t)(ka + 0) * O_DIM + n0 + l;
        const size_t bi1 = (size_t)(ka + 1) * O_DIM + n0 + l;
        v2f b1; b1.x = W1[bi0]; b1.y = W1[bi1];
        v2f b2; b2.x = W2[bi0]; b2.y = W2[bi1];

        c1 = __builtin_amdgcn_wmma_f32_16x16x4_f32(
                 /*neg_a=*/false, a, /*neg_b=*/false, b1,
                 /*c_mod=*/(short)0, c1, /*reuse_a=*/false, /*reuse_b=*/false);
        c2 = __builtin_amdgcn_wmma_f32_16x16x4_f32(
                 /*neg_a=*/false, a, /*neg_b=*/false, b2,
                 /*c_mod=*/(short)0, c2, /*reuse_a=*/false, /*reuse_b=*/false);
    }

    // C/D layout: VGPR j -> row (j + 8*half), col l
    const size_t obase = (size_t)(row0 + 8 * half) * O_DIM + n0 + l;
    if (row0 + 16 <= nRows) {
        // Uniform full-tile path: no EXEC masking, stores can be claused.
#pragma unroll
        for (int j = 0; j < 8; ++j) {
            const size_t o = obase + (size_t)j * O_DIM;
            Y1[o] = c1[j];
            Y2[o] = c2[j];
        }
    } else {
        // Tail tile only.
#pragma unroll
        for (int j = 0; j < 8; ++j) {
            const int mrow = row0 + j + 8 * half;
            if (mrow < nRows) {
                const size_t o = (size_t)mrow * O_DIM + n0 + l;
                Y1[o] = c1[j];
                Y2[o] = c2[j];
            }
        }
    }
}

// ---------------------------------------------------------------------------
// COO SpMM scatter: acc[row] += val * x[col]
// One wave (32 lanes) per edge; each lane covers 4 of the 128 columns.
// Gathers are full contiguous 512B rows (coalesced, L2-resident);
// scatter uses global_atomic_add_f32 (no return -> STOREcnt path).
// ---------------------------------------------------------------------------
__global__ void __launch_bounds__(256)
rgcn_spmm_atomic(const int*   __restrict__ rows,
                 const int*   __restrict__ cols,
                 const float* __restrict__ vals,
                 const float* __restrict__ Xp,
                 float*       __restrict__ acc,
                 int nEdges)
{
    const long gid = (long)blockIdx.x * blockDim.x + threadIdx.x;
    const int  e   = (int)(gid >> 5);
    if (e >= nEdges) return;
    const int lane = threadIdx.x & 31;

    const int   r = rows[e];
    const int   c = cols[e];
    const float v = vals[e];

    const float4 xv = *(const float4*)(Xp + (size_t)c * O_DIM + lane * 4);
    float* outp = acc + (size_t)r * O_DIM + lane * 4;
    atomicAdd(outp + 0, v * xv.x);
    atomicAdd(outp + 1, v * xv.y);
    atomicAdd(outp + 2, v * xv.z);
    atomicAdd(outp + 3, v * xv.w);
}

// ---------------------------------------------------------------------------
__global__ void __launch_bounds__(256)
rgcn_zero4(float4* __restrict__ p, long n4)
{
    const long i = (long)blockIdx.x * blockDim.x + threadIdx.x;
    if (i < n4) p[i] = make_float4(0.f, 0.f, 0.f, 0.f);
}

__global__ void __launch_bounds__(256)
rgcn_relu4(float4* __restrict__ p, long n4)
{
    const long i = (long)blockIdx.x * blockDim.x + threadIdx.x;
    if (i < n4) {
        float4 v = p[i];
        v.x = fmaxf(v.x, 0.f);
        v.y = fmaxf(v.y, 0.f);
        v.z = fmaxf(v.z, 0.f);
        v.w = fmaxf(v.w, 0.f);
        p[i] = v;
    }
}

// ---------------------------------------------------------------------------
extern "C" void kernel_launch(void* const* d_in, const int* in_sizes, int n_in,
                              void* d_out, int out_size, void* d_ws, size_t ws_size,
                              hipStream_t stream)
{
    const float* X  = (const float*)d_in[0];
    const int*   r1 = (const int*)  d_in[1];
    const int*   c1 = (const int*)  d_in[2];
    const float* v1 = (const float*)d_in[3];
    const int*   r2 = (const int*)  d_in[4];
    const int*   c2 = (const int*)  d_in[5];
    const float* v2 = (const float*)d_in[6];
    const float* W1 = (const float*)d_in[7];
    const float* W2 = (const float*)d_in[8];
    float* out = (float*)d_out;

    const int nRows = in_sizes[0] / D_DIM;   // 50000
    const int nE1   = in_sizes[1];           // 800000
    const int nE2   = in_sizes[4];           // 800000

    float* x1 = (float*)d_ws;                       // [nRows, O]
    float* x2 = x1 + (size_t)nRows * O_DIM;         // [nRows, O]  (51.2 MB total)

    // 1) zero the accumulator (d_out doubles as accumulator)
    const long n4 = (long)out_size / 4;
    rgcn_zero4<<<(int)((n4 + 255) / 256), 256, 0, stream>>>((float4*)out, n4);

    // 2) dual GEMM with fp32 WMMA
    const int rowTiles = (nRows + 15) / 16;
    rgcn_gemm_dual<<<rowTiles, 256, 0, stream>>>(X, W1, W2, x1, x2, nRows);

    // 3) sparse scatter-accumulate, both relations
    const long t1 = (long)nE1 * 32;
    rgcn_spmm_atomic<<<(int)((t1 + 255) / 256), 256, 0, stream>>>(r1, c1, v1, x1, out, nE1);
    const long t2 = (long)nE2 * 32;
    rgcn_spmm_atomic<<<(int)((t2 + 255) / 256), 256, 0, stream>>>(r2, c2, v2, x2, out, nE2);

    // 4) ReLU in place
    rgcn_relu4<<<(int)((n4 + 255) / 256), 256, 0, stream>>>((float4*)out, n4);
}